// GraphUnet_9534827397797
// MI455X (gfx1250) — compile-verified
//
#include <hip/hip_runtime.h>
#include <hip/hip_bf16.h>
#include <math.h>

typedef __attribute__((ext_vector_type(2))) float v2f;
typedef __attribute__((ext_vector_type(8))) float v8f;

// ---------------------------------------------------------------------------
// fp32 WMMA GEMM (narrow): C[M,N] = A[M,K] @ B[K,N], row-major.
// One wave per 16x16 C tile. Used for the N=16 GCN propagations.
// ---------------------------------------------------------------------------
#define GEMM_WPB 4   // waves per block (tiles along N)

__global__ __launch_bounds__(GEMM_WPB * 32)
void wmma_gemm_f32(const float* __restrict__ A, const float* __restrict__ B,
                   float* __restrict__ C, int M, int N, int K) {
    const int wave = threadIdx.x >> 5;
    const int lane = threadIdx.x & 31;
    const int lo   = lane & 15;    // row-of-A / col-of-B within tile
    const int sel  = lane >> 4;    // K-pair selector (0: k0..k1, 1: k2..k3)

    const int tileN = (blockIdx.x * GEMM_WPB + wave) << 4;
    const int tileM = blockIdx.y << 4;
    if (tileN >= N || tileM >= M) return;   // wave-uniform; EXEC stays all-ones

    const float* Arow = A + (size_t)(tileM + lo) * K;
    const float* Bcol = B + (size_t)(tileN + lo);

    v8f acc = {};
    for (int k0 = 0; k0 < K; k0 += 4) {
        const int ka = k0 + (sel << 1);
        v2f a, b;
        a.x = Arow[ka];
        a.y = Arow[ka + 1];
        b.x = Bcol[(size_t)ka * N];
        b.y = Bcol[(size_t)(ka + 1) * N];
        acc = __builtin_amdgcn_wmma_f32_16x16x4_f32(false, a, false, b,
                                                    (short)0, acc, false, false);
    }

    float* Cr = C + (size_t)tileM * N + tileN + lo;
#pragma unroll
    for (int r = 0; r < 8; ++r)
        Cr[(size_t)(r + (sel << 3)) * N] = acc[r];
}

// ---------------------------------------------------------------------------
// fp32 WMMA GEMM (register-blocked 2x2): 32x32 macro-tile per wave.
// Each A/B fragment is reused twice -> 8 FLOP/byte of fragment traffic.
// Requires M,N % 32 == 0, K % 4 == 0. Used for the big square A@A GEMMs.
// ---------------------------------------------------------------------------
__global__ __launch_bounds__(GEMM_WPB * 32)
void wmma_gemm_f32_2x2(const float* __restrict__ A, const float* __restrict__ B,
                       float* __restrict__ C, int M, int N, int K) {
    const int wave = threadIdx.x >> 5;
    const int lane = threadIdx.x & 31;
    const int lo   = lane & 15;
    const int sel  = lane >> 4;

    const int tileN = (blockIdx.x * GEMM_WPB + wave) << 5;   // 32-wide
    const int tileM = blockIdx.y << 5;                       // 32-tall
    if (tileN >= N || tileM >= M) return;   // wave-uniform

    const float* Arow0 = A + (size_t)(tileM + lo) * K;
    const float* Arow1 = Arow0 + (size_t)16 * K;
    const float* Bcol0 = B + (size_t)(tileN + lo);
    const float* Bcol1 = Bcol0 + 16;

    v8f acc00 = {}, acc01 = {}, acc10 = {}, acc11 = {};
    for (int k0 = 0; k0 < K; k0 += 4) {
        const int ka = k0 + (sel << 1);
        v2f a0, a1, b0, b1;
        a0.x = Arow0[ka];
        a0.y = Arow0[ka + 1];
        a1.x = Arow1[ka];
        a1.y = Arow1[ka + 1];
        b0.x = Bcol0[(size_t)ka * N];
        b0.y = Bcol0[(size_t)(ka + 1) * N];
        b1.x = Bcol1[(size_t)ka * N];
        b1.y = Bcol1[(size_t)(ka + 1) * N];
        acc00 = __builtin_amdgcn_wmma_f32_16x16x4_f32(false, a0, false, b0,
                                                      (short)0, acc00, false, false);
        acc01 = __builtin_amdgcn_wmma_f32_16x16x4_f32(false, a0, false, b1,
                                                      (short)0, acc01, false, false);
        acc10 = __builtin_amdgcn_wmma_f32_16x16x4_f32(false, a1, false, b0,
                                                      (short)0, acc10, false, false);
        acc11 = __builtin_amdgcn_wmma_f32_16x16x4_f32(false, a1, false, b1,
                                                      (short)0, acc11, false, false);
    }

    float* C00 = C + (size_t)tileM * N + tileN + lo;
#pragma unroll
    for (int r = 0; r < 8; ++r) {
        const size_t row = (size_t)(r + (sel << 3)) * N;
        C00[row]                     = acc00[r];
        C00[row + 16]                = acc01[r];
        C00[row + (size_t)16 * N]    = acc10[r];
        C00[row + (size_t)16 * N + 16] = acc11[r];
    }
}

// ---------------------------------------------------------------------------
// d[i] = 1/sqrt(rowsum(A) + 2)   (rowsum of A+2I)
// ---------------------------------------------------------------------------
__global__ void rowsum_dinv(const float* __restrict__ A, float* __restrict__ d, int n) {
    const int row = blockIdx.x;
    const float* a = A + (size_t)row * n;
    float s = 0.f;
    for (int j = threadIdx.x; j < n; j += blockDim.x) s += a[j];
    __shared__ float red[256];
    red[threadIdx.x] = s;
    __syncthreads();
    for (int off = blockDim.x >> 1; off > 0; off >>= 1) {
        if (threadIdx.x < off) red[threadIdx.x] += red[threadIdx.x + off];
        __syncthreads();
    }
    if (threadIdx.x == 0) d[row] = rsqrtf(red[0] + 2.0f);
}

// An[i,j] = (A[i,j] + 2*(i==j)) * d[i] * d[j]
__global__ void normalize_adj(const float* __restrict__ A, const float* __restrict__ d,
                              float* __restrict__ An, int n) {
    const int j = blockIdx.x * blockDim.x + threadIdx.x;
    const int i = blockIdx.y;
    if (j >= n) return;
    float v = A[(size_t)i * n + j] + ((i == j) ? 2.0f : 0.0f);
    An[(size_t)i * n + j] = v * d[i] * d[j];
}

// T[n x 16] = pad16( X[n x in_dim] @ W[in_dim x hid] )
__global__ void feat_xw(const float* __restrict__ X, const float* __restrict__ W,
                        float* __restrict__ T, int n, int in_dim, int hid) {
    const int t = blockIdx.x * blockDim.x + threadIdx.x;
    const int i = t >> 4, c = t & 15;
    if (i >= n) return;
    float s = 0.f;
    if (c < hid) {
        const float* x = X + (size_t)i * in_dim;
        for (int k = 0; k < in_dim; ++k) s += x[k] * W[k * hid + c];
    }
    T[t] = s;
}

// out[n x 3] = relu(Y[:, :3] + b) (+ skip)
__global__ void gcn_finish(const float* __restrict__ Y, const float* __restrict__ b,
                           const float* __restrict__ skip, float* __restrict__ out, int n) {
    const int t = blockIdx.x * blockDim.x + threadIdx.x;
    if (t >= n * 3) return;
    const int i = t / 3, c = t - i * 3;
    float v = Y[(size_t)i * 16 + c] + b[c];
    v = v > 0.f ? v : 0.f;
    if (skip) v += skip[t];
    out[t] = v;
}

// proj[i] = X[i,:] . pw + pb
__global__ void proj_kernel(const float* __restrict__ X, const float* __restrict__ pw,
                            const float* __restrict__ pb, float* __restrict__ proj, int n) {
    const int i = blockIdx.x * blockDim.x + threadIdx.x;
    if (i >= n) return;
    proj[i] = X[i * 3] * pw[0] + X[i * 3 + 1] * pw[1] + X[i * 3 + 2] * pw[2] + pb[0];
}

// sel[i] = 1 iff stable rank of proj[i] (desc) < k
__global__ void rank_sel(const float* __restrict__ proj, int* __restrict__ sel, int n, int k) {
    const int i = blockIdx.x * blockDim.x + threadIdx.x;
    if (i >= n) return;
    const float pi = proj[i];
    int r = 0;
    for (int j = 0; j < n; ++j) {
        const float pj = proj[j];
        r += (pj > pi) || (pj == pi && j < i);
    }
    sel[i] = (r < k) ? 1 : 0;
}

// compact selected indices in ascending index order; y[pos] = sigmoid(proj[i])
__global__ void compact_sel(const int* __restrict__ sel, const float* __restrict__ proj,
                            int* __restrict__ idx, float* __restrict__ y, int n) {
    const int i = blockIdx.x * blockDim.x + threadIdx.x;
    if (i >= n || !sel[i]) return;
    int p = 0;
    for (int j = 0; j < i; ++j) p += sel[j];
    idx[p] = i;
    y[p] = 1.0f / (1.0f + expf(-proj[i]));
}

// Ap[a,b] = A2[idx[a], idx[b]]
__global__ void gather_pool(const float* __restrict__ A2, const int* __restrict__ idx,
                            float* __restrict__ Ap, int n, int k) {
    const int b = blockIdx.x * blockDim.x + threadIdx.x;
    const int a = blockIdx.y;
    if (b >= k) return;
    Ap[(size_t)a * k + b] = A2[(size_t)idx[a] * n + idx[b]];
}

// Xp[a,c] = X[idx[a],c] * y[a]
__global__ void gather_feat(const float* __restrict__ X, const int* __restrict__ idx,
                            const float* __restrict__ y, float* __restrict__ Xp, int k) {
    const int t = blockIdx.x * blockDim.x + threadIdx.x;
    if (t >= k * 3) return;
    const int a = t / 3, c = t - a * 3;
    Xp[t] = X[idx[a] * 3 + c] * y[a];
}

__global__ void zero_f32(float* __restrict__ p, int n) {
    const int i = blockIdx.x * blockDim.x + threadIdx.x;
    if (i < n) p[i] = 0.f;
}

// Xu[idx[a],c] = Xp[a,c]
__global__ void scatter_feat(const float* __restrict__ Xp, const int* __restrict__ idx,
                             float* __restrict__ Xu, int k) {
    const int t = blockIdx.x * blockDim.x + threadIdx.x;
    if (t >= k * 3) return;
    const int a = t / 3, c = t - a * 3;
    Xu[idx[a] * 3 + c] = Xp[t];
}

// ---------------------------------------------------------------------------
extern "C" void kernel_launch(void* const* d_in, const int* in_sizes, int n_in,
                              void* d_out, int out_size, void* d_ws, size_t ws_size,
                              hipStream_t stream) {
    const float* g   = (const float*)d_in[0];
    const float* h   = (const float*)d_in[1];
    const float* W1  = (const float*)d_in[2];
    const float* b1  = (const float*)d_in[3];
    const float* W2  = (const float*)d_in[4];
    const float* b2  = (const float*)d_in[5];
    const float* Wb  = (const float*)d_in[6];
    const float* bb  = (const float*)d_in[7];
    const float* Wu1 = (const float*)d_in[8];
    const float* bu1 = (const float*)d_in[9];
    const float* Wu2 = (const float*)d_in[10];
    const float* bu2 = (const float*)d_in[11];
    const float* p1w = (const float*)d_in[12];
    const float* p1b = (const float*)d_in[13];
    const float* p2w = (const float*)d_in[14];
    const float* p2b = (const float*)d_in[15];
    float* out = (float*)d_out;
    (void)in_sizes; (void)n_in; (void)out_size; (void)ws_size;

    constexpr int N0 = 4096, P0 = 2048, P1 = 1024;

    // workspace layout (~177 MB of f32)
    float* ws = (float*)d_ws;
    size_t o = 0;
    auto alloc = [&](size_t cnt) { float* p = ws + o; o += cnt; return p; };
    float* An0 = alloc((size_t)N0 * N0);
    float* A2  = alloc((size_t)N0 * N0);   // reused for both pooling levels
    float* g1  = alloc((size_t)P0 * P0);
    float* An1 = alloc((size_t)P0 * P0);
    float* g2  = alloc((size_t)P1 * P1);
    float* An2 = alloc((size_t)P1 * P1);
    float* dv  = alloc(N0);
    float* T   = alloc((size_t)N0 * 16);
    float* Y   = alloc((size_t)N0 * 16);
    float* h1  = alloc(N0 * 3);
    float* h2  = alloc(P0 * 3);
    float* hb  = alloc(P1 * 3);
    float* h1p = alloc(P0 * 3);
    float* h2p = alloc(P1 * 3);
    float* hu1 = alloc(P0 * 3);
    float* hu2 = alloc(N0 * 3);
    float* pr  = alloc(N0);
    float* y1  = alloc(P0);
    float* y2  = alloc(P1);
    int* sel   = (int*)alloc(N0);
    int* idx1  = (int*)alloc(P0);
    int* idx2  = (int*)alloc(P1);

    // narrow GEMM (N == 16 propagations)
    auto gemm16 = [&](const float* A, const float* B, float* C, int M, int N, int K) {
        dim3 grid((N / 16 + GEMM_WPB - 1) / GEMM_WPB, M / 16);
        wmma_gemm_f32<<<grid, GEMM_WPB * 32, 0, stream>>>(A, B, C, M, N, K);
    };
    // register-blocked GEMM for big square products (dims % 32 == 0)
    auto gemm_big = [&](const float* A, const float* B, float* C, int M, int N, int K) {
        dim3 grid((N / 32 + GEMM_WPB - 1) / GEMM_WPB, M / 32);
        wmma_gemm_f32_2x2<<<grid, GEMM_WPB * 32, 0, stream>>>(A, B, C, M, N, K);
    };

    // ---- level 0: h1 = gcn(h, g, W1, b1) ----
    rowsum_dinv<<<N0, 256, 0, stream>>>(g, dv, N0);
    normalize_adj<<<dim3(N0 / 256, N0), 256, 0, stream>>>(g, dv, An0, N0);
    feat_xw<<<(N0 * 16 + 255) / 256, 256, 0, stream>>>(h, W1, T, N0, 128, 3);
    gemm16(An0, T, Y, N0, 16, N0);
    gcn_finish<<<(N0 * 3 + 255) / 256, 256, 0, stream>>>(Y, b1, nullptr, h1, N0);

    // ---- pool 1: A2 = g@g (dominant GEMM), top-2048 ----
    gemm_big(g, g, A2, N0, N0, N0);
    proj_kernel<<<(N0 + 255) / 256, 256, 0, stream>>>(h1, p1w, p1b, pr, N0);
    rank_sel<<<(N0 + 255) / 256, 256, 0, stream>>>(pr, sel, N0, P0);
    compact_sel<<<(N0 + 255) / 256, 256, 0, stream>>>(sel, pr, idx1, y1, N0);
    gather_pool<<<dim3(P0 / 256, P0), 256, 0, stream>>>(A2, idx1, g1, N0, P0);
    gather_feat<<<(P0 * 3 + 255) / 256, 256, 0, stream>>>(h1, idx1, y1, h1p, P0);

    // ---- level 1: h2 = gcn(h1p, g1, W2, b2) ----
    rowsum_dinv<<<P0, 256, 0, stream>>>(g1, dv, P0);
    normalize_adj<<<dim3(P0 / 256, P0), 256, 0, stream>>>(g1, dv, An1, P0);
    feat_xw<<<(P0 * 16 + 255) / 256, 256, 0, stream>>>(h1p, W2, T, P0, 3, 3);
    gemm16(An1, T, Y, P0, 16, P0);
    gcn_finish<<<(P0 * 3 + 255) / 256, 256, 0, stream>>>(Y, b2, nullptr, h2, P0);

    // ---- pool 2: A2 = g1@g1, top-1024 ----
    gemm_big(g1, g1, A2, P0, P0, P0);
    proj_kernel<<<(P0 + 255) / 256, 256, 0, stream>>>(h2, p2w, p2b, pr, P0);
    rank_sel<<<(P0 + 255) / 256, 256, 0, stream>>>(pr, sel, P0, P1);
    compact_sel<<<(P0 + 255) / 256, 256, 0, stream>>>(sel, pr, idx2, y2, P0);
    gather_pool<<<dim3(P1 / 256, P1), 256, 0, stream>>>(A2, idx2, g2, P0, P1);
    gather_feat<<<(P1 * 3 + 255) / 256, 256, 0, stream>>>(h2, idx2, y2, h2p, P1);

    // ---- bottom: hb = gcn(h2p, g2, Wb, bb) ----
    rowsum_dinv<<<P1, 256, 0, stream>>>(g2, dv, P1);
    normalize_adj<<<dim3(P1 / 256, P1), 256, 0, stream>>>(g2, dv, An2, P1);
    feat_xw<<<(P1 * 16 + 255) / 256, 256, 0, stream>>>(h2p, Wb, T, P1, 3, 3);
    gemm16(An2, T, Y, P1, 16, P1);
    gcn_finish<<<(P1 * 3 + 255) / 256, 256, 0, stream>>>(Y, bb, nullptr, hb, P1);

    // ---- up 1: unpool(idx2) -> gcn(An1, Wu1) + h2 ----
    zero_f32<<<(P0 * 3 + 255) / 256, 256, 0, stream>>>(hu1, P0 * 3);
    scatter_feat<<<(P1 * 3 + 255) / 256, 256, 0, stream>>>(hb, idx2, hu1, P1);
    feat_xw<<<(P0 * 16 + 255) / 256, 256, 0, stream>>>(hu1, Wu1, T, P0, 3, 3);
    gemm16(An1, T, Y, P0, 16, P0);
    gcn_finish<<<(P0 * 3 + 255) / 256, 256, 0, stream>>>(Y, bu1, h2, hu1, P0);

    // ---- up 2: unpool(idx1) -> gcn(An0, Wu2) + h1 -> d_out ----
    zero_f32<<<(N0 * 3 + 255) / 256, 256, 0, stream>>>(hu2, N0 * 3);
    scatter_feat<<<(P0 * 3 + 255) / 256, 256, 0, stream>>>(hu1, idx1, hu2, P0);
    feat_xw<<<(N0 * 16 + 255) / 256, 256, 0, stream>>>(hu2, Wu2, T, N0, 3, 3);
    gemm16(An0, T, Y, N0, 16, N0);
    gcn_finish<<<(N0 * 3 + 255) / 256, 256, 0, stream>>>(Y, bu2, h1, out, N0);

    // ---- second output: ga = g ----
    hipMemcpyAsync(out + (size_t)N0 * 3, g, (size_t)N0 * N0 * sizeof(float),
                   hipMemcpyDeviceToDevice, stream);
}